// Polarization_11149735100681
// MI455X (gfx1250) — compile-verified
//
#include <hip/hip_runtime.h>
#include <stdint.h>

#define NUM_GRAPHS     1024
#define TILE           1024        // nodes per staged tile
#define BLOCK_THREADS  256
#define NODES_PER_THR  4           // TILE / BLOCK_THREADS
#define NBLOCKS        2048

typedef unsigned int v4u __attribute__((ext_vector_type(4)));
typedef int          v4i __attribute__((ext_vector_type(4)));
typedef int          v8i __attribute__((ext_vector_type(8)));
typedef float        v2f __attribute__((ext_vector_type(2)));
typedef float        v8f __attribute__((ext_vector_type(8)));

// ---------------------------------------------------------------------------
// 1-D TDM copy: `tile_elems` dwords from gptr -> LDS byte address lds_addr.
// tensor_dim0 is set to the remaining dwords in the buffer so the TDM
// zero-fills (instead of faulting) on the tail tile.
// D# layout per CDNA5 ISA.
// ---------------------------------------------------------------------------
__device__ __forceinline__ void tdm_load_1d(unsigned lds_addr,
                                            const void* gptr,
                                            unsigned tile_elems,
                                            unsigned long long remaining_elems)
{
  unsigned long long ga = (unsigned long long)gptr;

  v4u g0;
  g0.x = 1u;                                            // count=1, user descriptor
  g0.y = lds_addr;                                      // LDS byte address
  g0.z = (unsigned)(ga & 0xFFFFFFFFu);                  // global_addr[31:0]
  g0.w = (unsigned)((ga >> 32) & 0x01FFFFFFu) | (2u << 30); // addr[56:32] | type=2

  unsigned long long d0ll = remaining_elems;
  if (d0ll > 0x7FFFFFFFull) d0ll = 0x7FFFFFFFull;
  unsigned d0 = (unsigned)d0ll;

  v8i g1;
  g1[0] = (int)(2u << 16);                    // data_size=2 -> 4-byte elements
  g1[1] = (int)((d0 & 0xFFFFu) << 16);        // tensor_dim0[15:0]
  g1[2] = (int)((d0 >> 16) & 0xFFFFu);        // tensor_dim0[31:16]; tensor_dim1=0
  g1[3] = (int)((tile_elems & 0xFFFFu) << 16);// tile_dim0
  g1[4] = 0;                                  // tile_dim1=0 (1-D), tile_dim2=0
  g1[5] = 0; g1[6] = 0; g1[7] = 0;            // strides unused for 1-D

  v4i gz = {0, 0, 0, 0};
#if defined(__clang_major__) && (__clang_major__ >= 23)
  v8i gz8 = {0, 0, 0, 0, 0, 0, 0, 0};
  __builtin_amdgcn_tensor_load_to_lds(g0, g1, gz, gz, gz8, 0);
#else
  __builtin_amdgcn_tensor_load_to_lds(g0, g1, gz, gz, 0);
#endif
}

// ---------------------------------------------------------------------------
// Main streaming segmented reduction (defined first so the disasm snippet
// shows the tensor_load_to_lds pipeline).
//   acc[g*7 + 0..2] = sum_{i in g} q_i * r_i
//   acc[g*7 + 3..5] = sum_{i in g} r_i
//   acc[g*7 + 6]    = sum_{i in g} q_i
// positions/q are TDM-DMA'd into double-buffered LDS; batch is read with
// coalesced global loads (values < 1024, int64 storage).
// ---------------------------------------------------------------------------
__global__ __launch_bounds__(BLOCK_THREADS)
void polar_reduce_kernel(const float* __restrict__ pos,
                         const float* __restrict__ q,
                         const long long* __restrict__ batch,
                         float* __restrict__ acc,
                         int n, int tiles_per_block, int total_tiles)
{
  __shared__ float s_pos[2][TILE * 3];          // 2 x 12 KB
  __shared__ float s_q[2][TILE];                // 2 x  4 KB
  __shared__ float s_bins[NUM_GRAPHS * 7];      // 28 KB

  const int tid = threadIdx.x;
  for (int k = tid; k < NUM_GRAPHS * 7; k += BLOCK_THREADS) s_bins[k] = 0.0f;

  const int tile0   = blockIdx.x * tiles_per_block;
  int tileEnd       = tile0 + tiles_per_block;
  if (tileEnd > total_tiles) tileEnd = total_tiles;
  const int numT    = tileEnd - tile0;
  if (numT <= 0) return;                        // uniform over the block

  const bool issuer = (tid < 32);               // wave 0 only (wave-uniform)

  auto stage = [&](int t, int buf) {
    long long start = (long long)(tile0 + t) * TILE;
    long long remN  = (long long)n - start; if (remN < 0) remN = 0;
    unsigned lds_pos = (unsigned)(unsigned long long)(&s_pos[buf][0]);
    unsigned lds_q   = (unsigned)(unsigned long long)(&s_q[buf][0]);
    tdm_load_1d(lds_pos, pos + start * 3, TILE * 3, (unsigned long long)remN * 3ull);
    tdm_load_1d(lds_q,   q + start,       TILE,     (unsigned long long)remN);
  };

  if (issuer) stage(0, 0);

  float a0 = 0.f, a1 = 0.f, a2 = 0.f, a3 = 0.f, a4 = 0.f, a5 = 0.f, a6 = 0.f;
  int cur = -1;

  auto flushRun = [&]() {
    atomicAdd(&s_bins[cur * 7 + 0], a0);
    atomicAdd(&s_bins[cur * 7 + 1], a1);
    atomicAdd(&s_bins[cur * 7 + 2], a2);
    atomicAdd(&s_bins[cur * 7 + 3], a3);
    atomicAdd(&s_bins[cur * 7 + 4], a4);
    atomicAdd(&s_bins[cur * 7 + 5], a5);
    atomicAdd(&s_bins[cur * 7 + 6], a6);
    a0 = a1 = a2 = a3 = a4 = a5 = a6 = 0.f;
  };

  for (int t = 0; t < numT; ++t) {
    if (t + 1 < numT) {
      if (issuer) stage(t + 1, (t + 1) & 1);
      __builtin_amdgcn_s_wait_tensorcnt(2);     // tile t done; t+1 in flight
    } else {
      __builtin_amdgcn_s_wait_tensorcnt(0);     // last tile: drain
    }
    __syncthreads();                            // tile t visible to all waves

    const int buf = t & 1;
    const long long base = (long long)(tile0 + t) * TILE;
#pragma unroll
    for (int j = 0; j < NODES_PER_THR; ++j) {
      const int l = tid * NODES_PER_THR + j;    // thread-contiguous run
      const long long i = base + l;
      if (i < n) {
        const int b = (int)batch[i];
        if (b != cur) {
          if (cur >= 0) flushRun();
          cur = b;
        }
        const float qv = s_q[buf][l];
        const float x  = s_pos[buf][l * 3 + 0];
        const float y  = s_pos[buf][l * 3 + 1];
        const float z  = s_pos[buf][l * 3 + 2];
        a0 += qv * x; a1 += qv * y; a2 += qv * z;
        a3 += x;      a4 += y;      a5 += z;
        a6 += qv;
      }
    }
    __syncthreads();                            // buf may be overwritten next iter
  }
  if (cur >= 0) flushRun();
  __syncthreads();

  // Flush only the graph range this block's contiguous chunk touched.
  long long first = (long long)tile0 * TILE;
  long long last  = (long long)tileEnd * TILE;
  if (last > n) last = n;
  last -= 1;
  if (first <= last) {
    const int gmin = (int)batch[first];
    const int gmax = (int)batch[last];
    const int span = (gmax - gmin + 1) * 7;
    for (int k = tid; k < span; k += BLOCK_THREADS) {
      const int g = gmin + k / 7;
      const int c = k % 7;
      const float v = s_bins[g * 7 + c];
      if (v != 0.0f) atomicAdd(&acc[g * 7 + c], v);
    }
  }
}

// ---------------------------------------------------------------------------
// finalize: mu = (sum_g acc[g][6]) / n ;  pol[g] = acc[g][0..2] - mu*acc[g][3..5]
//
// The 1024-wide sum-of-sums uses V_WMMA_F32_16X16X4_F32 with an all-ones
// A matrix: D += ones(16x4) x B(4x16) makes every row of D the column sums
// of B, so the result is independent of the exact A/B VGPR striping (each of
// the 1024 values occupies exactly one of the 64 B slots per tile, and we
// sum all 16 columns at the end). Exact f32, one wave, 16 accumulating WMMAs.
// ---------------------------------------------------------------------------
__global__ __launch_bounds__(1024)
void finalize_kernel(const float* __restrict__ acc, float* __restrict__ out,
                     float inv_n)
{
  __shared__ float red[NUM_GRAPHS];
  __shared__ float s_tot;
  const int g = threadIdx.x;
  red[g] = acc[g * 7 + 6];
  __syncthreads();

  if (threadIdx.x < 32) {                       // wave 0, EXEC all ones
    const int lane = threadIdx.x;
    v2f a; a.x = 1.0f; a.y = 1.0f;              // A = ones (layout-independent)
    v8f d = {0.f, 0.f, 0.f, 0.f, 0.f, 0.f, 0.f, 0.f};
#pragma unroll
    for (int t = 0; t < NUM_GRAPHS / 64; ++t) { // 16 tiles x 64 values
      v2f b;
      b.x = red[t * 64 + lane];
      b.y = red[t * 64 + 32 + lane];
      d = __builtin_amdgcn_wmma_f32_16x16x4_f32(
          /*neg_a=*/false, a, /*neg_b=*/false, b,
          /*c_mod=*/(short)0, d, /*reuse_a=*/false, /*reuse_b=*/false);
    }
    red[lane] = d[0];                           // row 0 of D (rows identical)
    if (lane == 0) {
      float tot = 0.f;
      for (int k = 0; k < 16; ++k) tot += red[k];  // sum the 16 column sums
      s_tot = tot;
    }
  }
  __syncthreads();

  const float mu = s_tot * inv_n;
  out[g * 3 + 0] = acc[g * 7 + 0] - mu * acc[g * 7 + 3];
  out[g * 3 + 1] = acc[g * 7 + 1] - mu * acc[g * 7 + 4];
  out[g * 3 + 2] = acc[g * 7 + 2] - mu * acc[g * 7 + 5];
}

__global__ void zero_kernel(float* __restrict__ p, int n)
{
  int i = blockIdx.x * blockDim.x + threadIdx.x;
  if (i < n) p[i] = 0.0f;
}

extern "C" void kernel_launch(void* const* d_in, const int* in_sizes, int n_in,
                              void* d_out, int out_size, void* d_ws, size_t ws_size,
                              hipStream_t stream)
{
  const float*     pos   = (const float*)d_in[0];
  const float*     q     = (const float*)d_in[1];
  const long long* batch = (const long long*)d_in[2];
  float*           out   = (float*)d_out;
  float*           acc   = (float*)d_ws;        // NUM_GRAPHS*7 floats

  const int n    = in_sizes[1];                 // q element count == N_NODES
  const int accN = NUM_GRAPHS * 7;

  zero_kernel<<<(accN + 255) / 256, 256, 0, stream>>>(acc, accN);

  const int total_tiles = (n + TILE - 1) / TILE;
  const int tpb         = (total_tiles + NBLOCKS - 1) / NBLOCKS;
  polar_reduce_kernel<<<NBLOCKS, BLOCK_THREADS, 0, stream>>>(
      pos, q, batch, acc, n, tpb, total_tiles);

  finalize_kernel<<<1, NUM_GRAPHS, 0, stream>>>(acc, out, 1.0f / (float)n);
}